// TwoWordPSDProbe_9156870275539
// MI455X (gfx1250) — compile-verified
//
#include <hip/hip_runtime.h>

typedef __attribute__((ext_vector_type(2))) float v2f;
typedef __attribute__((ext_vector_type(4))) float v4f;
typedef __attribute__((ext_vector_type(8))) float v8f;

#define S_DIM 1024
#define D_DIM 1024
#define R_DIM 64
#define NBATCH 4
#define M_TOT (NBATCH * S_DIM)   // 4096 total rows of T

// ---------------------------------------------------------------------------
// Kernel 1: T = batch @ proj   (4096 x 1024) @ (1024 x 64) -> (4096 x 64)
// One wave per 16x16 tile of T. 256 K-steps of v_wmma_f32_16x16x4_f32.
// grid: 256 i-tiles * 4 n-tiles = 1024 waves = 128 blocks of 256 threads.
// ---------------------------------------------------------------------------
__global__ __launch_bounds__(256) void proj_gemm_wmma(
    const float* __restrict__ batch, const float* __restrict__ proj,
    float* __restrict__ T)
{
  const int wave = (blockIdx.x * blockDim.x + threadIdx.x) >> 5;
  const int lane = threadIdx.x & 31;
  const int half = lane >> 4;     // 0: lanes 0-15, 1: lanes 16-31
  const int lh   = lane & 15;

  const int itile = wave >> 2;    // 0..255
  const int ntile = wave & 3;     // 0..3
  const int i0 = itile * 16;
  const int n0 = ntile * 16;

  // A lane layout (32-bit 16x4): lane lh holds row M=i0+lh; V0=K(2*half), V1=K(2*half+1)
  const float* arow = batch + (size_t)(i0 + lh) * D_DIM + 2 * half;
  // B lane layout (32-bit 4x16): lane lh holds col N=n0+lh; V0=row K(2*half), V1=row K(2*half+1)
  const float* bcol = proj + (size_t)(2 * half) * R_DIM + n0 + lh;

  v8f c = {0.f, 0.f, 0.f, 0.f, 0.f, 0.f, 0.f, 0.f};
#pragma unroll 4
  for (int kk = 0; kk < D_DIM; kk += 4) {
    v2f a = *(const v2f*)(arow + kk);          // 8B-aligned: kk even, row base even
    v2f b;
    b.x = bcol[(size_t)kk * R_DIM];
    b.y = bcol[(size_t)(kk + 1) * R_DIM];
    c = __builtin_amdgcn_wmma_f32_16x16x4_f32(
        /*neg_a=*/false, a, /*neg_b=*/false, b,
        /*c_mod=*/(short)0, c, /*reuse_a=*/false, /*reuse_b=*/false);
  }

  // C layout: VGPR v -> row (v + 8*half), col lh
#pragma unroll
  for (int v = 0; v < 8; ++v) {
    const int row = i0 + v + 8 * half;
    T[(size_t)row * R_DIM + n0 + lh] = c[v];
  }
}

// ---------------------------------------------------------------------------
// Kernel 2: nrm[r] = sum_k T[r][k]^2   (4096 rows, T is L2-resident, 1 MB)
// ---------------------------------------------------------------------------
__global__ __launch_bounds__(256) void row_norms(
    const float* __restrict__ T, float* __restrict__ nrm)
{
  const int row = blockIdx.x * blockDim.x + threadIdx.x;
  if (row >= M_TOT) return;
  const v4f* p = (const v4f*)(T + (size_t)row * R_DIM);
  float s = 0.f;
#pragma unroll
  for (int k = 0; k < R_DIM / 4; ++k) {
    v4f v = p[k];
    s += v.x * v.x + v.y * v.y + v.z * v.z + v.w * v.w;
  }
  nrm[row] = s;
}

// ---------------------------------------------------------------------------
// Kernel 3: out[b,i,j] = nrm[i] + nrm[j] - 2 * (T_b T_b^T)[i,j]
// One wave per 16x64 output strip: A tile hoisted to registers (16 float2),
// 4 j-tiles x 16 WMMAs each, fp32 epilogue, coalesced 16 MB output stream.
// grid: 4 batches * 64 i-tiles * 16 j-groups = 4096 waves = 512 blocks.
// ---------------------------------------------------------------------------
__global__ __launch_bounds__(256) void pdist_wmma(
    const float* __restrict__ T, const float* __restrict__ nrm,
    float* __restrict__ out)
{
  const int wave = (blockIdx.x * blockDim.x + threadIdx.x) >> 5;
  const int lane = threadIdx.x & 31;
  const int half = lane >> 4;
  const int lh   = lane & 15;

  const int b     = wave >> 10;        // / (64*16)
  const int rem   = wave & 1023;
  const int itile = rem >> 4;          // 0..63
  const int jgrp  = rem & 15;          // 0..15
  const int i0    = itile * 16;
  const int j0b   = jgrp * 64;

  const size_t rowbase = (size_t)b * S_DIM;

  // Hoist A: rows i0..i0+15, all K=64 (16 K-steps of 4)
  v2f areg[16];
  const float* Ti = T + (rowbase + i0 + lh) * R_DIM + 2 * half;
#pragma unroll
  for (int k = 0; k < 16; ++k) areg[k] = *(const v2f*)(Ti + 4 * k);

  float ni[8];
#pragma unroll
  for (int v = 0; v < 8; ++v) ni[v] = nrm[rowbase + i0 + v + 8 * half];

  float* outb = out + (size_t)b * S_DIM * S_DIM;

#pragma unroll
  for (int jt = 0; jt < 4; ++jt) {
    const int j0 = j0b + jt * 16;
    const float nj = nrm[rowbase + j0 + lh];
    const float* Tj = T + (rowbase + j0 + lh) * R_DIM + 2 * half;

    v8f c = {0.f, 0.f, 0.f, 0.f, 0.f, 0.f, 0.f, 0.f};
#pragma unroll
    for (int k = 0; k < 16; ++k) {
      v2f bv = *(const v2f*)(Tj + 4 * k);  // B[k][n] = T[j0+n][k] via lane layout
      c = __builtin_amdgcn_wmma_f32_16x16x4_f32(
          false, areg[k], false, bv, (short)0, c, false, false);
    }

#pragma unroll
    for (int v = 0; v < 8; ++v) {
      const int i = i0 + v + 8 * half;
      outb[(size_t)i * S_DIM + j0 + lh] = fmaf(-2.f, c[v], ni[v] + nj);
    }
  }
}

// ---------------------------------------------------------------------------
extern "C" void kernel_launch(void* const* d_in, const int* in_sizes, int n_in,
                              void* d_out, int out_size, void* d_ws, size_t ws_size,
                              hipStream_t stream)
{
  const float* batch = (const float*)d_in[0];  // (4, 1024, 1024) fp32
  const float* proj  = (const float*)d_in[1];  // (1024, 64)      fp32
  float* out = (float*)d_out;                  // (4, 1024, 1024) fp32

  float* T   = (float*)d_ws;                   // 4096*64 floats  (1 MB)
  float* nrm = T + (size_t)M_TOT * R_DIM;      // 4096 floats     (16 KB)

  proj_gemm_wmma<<<128, 256, 0, stream>>>(batch, proj, T);
  row_norms<<<16, 256, 0, stream>>>(T, nrm);
  pdist_wmma<<<512, 256, 0, stream>>>(T, nrm, out);
}